// Layer_46488726012135
// MI455X (gfx1250) — compile-verified
//
#include <hip/hip_runtime.h>
#include <math.h>

typedef float v2f __attribute__((ext_vector_type(2)));
typedef float v8f __attribute__((ext_vector_type(8)));

// ---------------------------------------------------------------------------
// Phase 1: gate projections via V_WMMA_F32_16X16X4_F32.
//   D(16x16) = A(16x4) x B(4x16) accumulated over K=64 in 16 steps.
//   A row 0 = w_dst slice, A row 1 = w_src slice, rows 2..15 = 0 (masked).
//   B column n = h[nodeBase+n][k..k+3].
//   Result: D[0][n] = s_dst(node n), D[1][n] = s_src(node n).
// VGPR layouts per CDNA5 ISA 7.12.2:
//   A: lane l -> (M=l%16, K=2*(l/16)) in v0, K+1 in v1   => one float2 load
//   B: v0 holds rows K={0,2}, v1 holds rows K={1,3}, N=lane%16 => float2 load
//   C/D f32 16x16: VGPR r, lanes 0-15 -> (M=r, N=lane)
// A fragments are tile-invariant: preloaded once. B fragments are preloaded
// per tile so all 16 loads are in flight before the WMMA chain starts.
// ---------------------------------------------------------------------------
__global__ void fagcn_gate_wmma(const float* __restrict__ h,
                                const float* __restrict__ gate_w,
                                float2* __restrict__ s,
                                int nNodes, int nTiles) {
    const int lane          = threadIdx.x & 31;
    const int waveInBlock   = threadIdx.x >> 5;
    const int wavesPerBlock = blockDim.x >> 5;
    const int row  = lane & 15;  // A: M index, B: N index
    const int half = lane >> 4;  // 0 -> K pair {0,1}, 1 -> K pair {2,3}

    // ---- Preload A fragments (branchless, loop- and tile-invariant) ----
    v2f afrag[16];
    {
        const float msk = (row < 2) ? 1.0f : 0.0f;      // rows >= 2 are zero
        const float* wp = gate_w + (row & 1) * 64 + half * 2; // always valid addr
#pragma unroll
        for (int kk = 0; kk < 16; ++kk) {
            afrag[kk].x = wp[kk * 4]     * msk;
            afrag[kk].y = wp[kk * 4 + 1] * msk;
        }
    }

    for (int tile = blockIdx.x * wavesPerBlock + waveInBlock;
         tile < nTiles;
         tile += gridDim.x * wavesPerBlock) {
        const int nodeBase = tile * 16;
        int nodeLd = nodeBase + row;
        if (nodeLd >= nNodes) nodeLd = nNodes - 1;       // tail clamp (loads only)
        const float* hrow = h + (size_t)nodeLd * 64 + half * 2;

        // ---- Preload all 16 B fragments (loads issued back-to-back) ----
        v2f bfrag[16];
#pragma unroll
        for (int kk = 0; kk < 16; ++kk) {
            bfrag[kk].x = hrow[kk * 4];
            bfrag[kk].y = hrow[kk * 4 + 1];
        }

        // ---- Uninterrupted 16-step WMMA accumulation chain ----
        v8f c = {};
#pragma unroll
        for (int kk = 0; kk < 16; ++kk) {
            c = __builtin_amdgcn_wmma_f32_16x16x4_f32(
                    /*neg_a=*/false, afrag[kk], /*neg_b=*/false, bfrag[kk],
                    /*c_mod=*/(short)0, c, /*reuse_a=*/false, /*reuse_b=*/false);
        }

        if (half == 0 && (nodeBase + row) < nNodes) {
            // c[0] = s_dst, c[1] = s_src for node (nodeBase+row)
            s[nodeBase + row] = make_float2(c[0], c[1]);
        }
    }
}

// ---------------------------------------------------------------------------
// Phase 2: per-edge gate + gather + scatter-add (the bandwidth-bound phase).
// One wave handles 32 edges: each lane computes one edge's scalar weight
// (coalesced metadata loads), then the wave iterates over the 32 edges,
// broadcasting (src, dst, w) to SGPRs via v_readlane so the feature
// gather/scatter uses the scalar-base addressing path. All 32 lanes move
// each edge's 64 features as float2 loads + 2 no-return f32 atomics.
// h (25.6MB) and z (25.6MB) are L2-resident on MI455X (192MB L2).
// ---------------------------------------------------------------------------
__global__ void fagcn_edge_agg(const float* __restrict__ h,
                               const float* __restrict__ deg,
                               const float2* __restrict__ s,
                               const float* __restrict__ gate_b,
                               const int* __restrict__ src,
                               const int* __restrict__ dst,
                               float* __restrict__ z,
                               int nEdges) {
    const int lane   = threadIdx.x & 31;
    const int wid    = (blockIdx.x * blockDim.x + threadIdx.x) >> 5;
    const int nWaves = (gridDim.x * blockDim.x) >> 5;
    const float bias = gate_b[0];
    const int laneOff = lane * 2;

    for (int base = wid * 32; base < nEdges; base += nWaves * 32) {
        const int e      = base + lane;
        const bool valid = (e < nEdges);
        int sv = 0, dv = 0;
        float w = 0.0f;
        if (valid) {
            sv = src[e];
            dv = dst[e];
            const float g = tanhf(s[dv].x + s[sv].y + bias);
            w = deg[dv] * deg[sv] * g;
        }
        const int wbits = __float_as_int(w);

        for (int j = 0; j < 32; ++j) {
            const float wj = __int_as_float(__builtin_amdgcn_readlane(wbits, j));
            if (wj == 0.0f) continue;                 // scalar skip (tail / exact zeros)
            const int svj = __builtin_amdgcn_readlane(sv, j);
            const int dvj = __builtin_amdgcn_readlane(dv, j);
            const float2 hv = *(const float2*)(h + (size_t)svj * 64 + laneOff);
            float* zp = z + (size_t)dvj * 64 + laneOff;
            atomicAdd(zp,     hv.x * wj);
            atomicAdd(zp + 1, hv.y * wj);
        }
    }
}

extern "C" void kernel_launch(void* const* d_in, const int* in_sizes, int n_in,
                              void* d_out, int out_size, void* d_ws, size_t ws_size,
                              hipStream_t stream) {
    (void)n_in; (void)ws_size;
    const float* h      = (const float*)d_in[0];   // [N,64] f32
    const float* deg    = (const float*)d_in[1];   // [N]    f32
    const float* gate_w = (const float*)d_in[2];   // [1,128] f32
    const float* gate_b = (const float*)d_in[3];   // [1]    f32
    const int*   src    = (const int*)d_in[4];     // [E]    i32
    const int*   dst    = (const int*)d_in[5];     // [E]    i32
    float*       z      = (float*)d_out;           // [N,64] f32

    const int N = in_sizes[1];
    const int E = in_sizes[4];

    float2* s = (float2*)d_ws;                     // [N] (s_dst, s_src)

    // z = 0 (graph-capture-safe async memset)
    hipMemsetAsync(d_out, 0, (size_t)out_size * sizeof(float), stream);

    // Phase 1: gate projections via WMMA
    {
        const int nTiles        = (N + 15) / 16;
        const int wavesPerBlock = 8;               // 256 threads = 8 wave32
        int blocks = (nTiles + wavesPerBlock - 1) / wavesPerBlock;
        if (blocks > 4096) blocks = 4096;
        fagcn_gate_wmma<<<blocks, 256, 0, stream>>>(h, gate_w, s, N, nTiles);
    }

    // Phase 2: edge gate + message aggregation
    {
        const int blocks = 2048;                   // 16384 waves, grid-stride
        fagcn_edge_agg<<<blocks, 256, 0, stream>>>(h, deg, s, gate_b, src, dst, z, E);
    }
}